// Integrator_62577673502887
// MI455X (gfx1250) — compile-verified
//
#include <hip/hip_runtime.h>

typedef _Float16 half_t;
typedef _Float16 v8h __attribute__((ext_vector_type(8)));
typedef float    v4f __attribute__((ext_vector_type(4)));
typedef unsigned int u32;

#define TPB 256
#define MAX_WEIGHT 500.0f
#define TRUNC_VALUE 0.1f

// ---------------------------------------------------------------------------
// Kernel 1: zero the accumulator scratch (uint4 = 16B stores, fully coalesced)
// ---------------------------------------------------------------------------
__global__ void zero_ws_kernel(uint4* __restrict__ ws, long long n_vec4) {
    long long i = (long long)blockIdx.x * blockDim.x + threadIdx.x;
    if (i < n_vec4) {
        uint4 z; z.x = 0u; z.y = 0u; z.z = 0u; z.w = 0u;
        ws[i] = z;
    }
}

// ---------------------------------------------------------------------------
// Kernel 2: surface-update scatter (f32 atomics, no-return -> STOREcnt path)
// ---------------------------------------------------------------------------
__global__ void scatter_surface_kernel(const float* __restrict__ uv,
                                       const float* __restrict__ uf,
                                       const int*   __restrict__ uidx,
                                       const float* __restrict__ uw,
                                       float* __restrict__ w_sum,
                                       float* __restrict__ u_sum,
                                       float* __restrict__ feat_sum,
                                       u32*   __restrict__ touched,
                                       int N) {
    int i = blockIdx.x * blockDim.x + threadIdx.x;
    if (i >= N) return;
    int x = uidx[3 * i + 0];
    int y = uidx[3 * i + 1];
    int z = uidx[3 * i + 2];
    // get_index_mask: invalid points contribute exactly nothing
    if ((unsigned)x >= 256u || (unsigned)y >= 256u || (unsigned)z >= 256u) return;
    int lin = (x << 16) | (y << 8) | z;

    float w = uw[i];
    atomicOr(&touched[lin >> 5], 1u << (lin & 31));
    atomicAdd(&w_sum[lin], w);
    atomicAdd(&u_sum[lin], w * uv[i]);

    const v4f* fp = (const v4f*)(uf + (size_t)i * 8);
    v4f f0 = fp[0];
    v4f f1 = fp[1];
    float* fs = feat_sum + (size_t)lin * 8;
    atomicAdd(fs + 0, w * f0.x);
    atomicAdd(fs + 1, w * f0.y);
    atomicAdd(fs + 2, w * f0.z);
    atomicAdd(fs + 3, w * f0.w);
    atomicAdd(fs + 4, w * f1.x);
    atomicAdd(fs + 5, w * f1.y);
    atomicAdd(fs + 6, w * f1.z);
    atomicAdd(fs + 7, w * f1.w);
}

// ---------------------------------------------------------------------------
// Kernel 3: empty-space voting scatter
// ---------------------------------------------------------------------------
__global__ void scatter_empty_kernel(const int*   __restrict__ uidx,
                                     const float* __restrict__ uwe,
                                     float* __restrict__ we_sum,
                                     u32*   __restrict__ touchedE,
                                     int N) {
    int i = blockIdx.x * blockDim.x + threadIdx.x;
    if (i >= N) return;
    int x = uidx[3 * i + 0];
    int y = uidx[3 * i + 1];
    int z = uidx[3 * i + 2];
    if ((unsigned)x >= 256u || (unsigned)y >= 256u || (unsigned)z >= 256u) return;
    int lin = (x << 16) | (y << 8) | z;
    atomicOr(&touchedE[lin >> 5], 1u << (lin & 31));
    atomicAdd(&we_sum[lin], uwe[i]);
}

// ---------------------------------------------------------------------------
// Kernel 4: per-voxel finalize. One thread per voxel, exact grid (full EXEC).
// The 32B/voxel feat_sum row is staged into LDS with the CDNA5 async-copy
// engine (global_load_async_to_lds_b128, ASYNCcnt) so the scalar v/w blend
// overlaps the feature-row fetch; s_wait_asynccnt 0 before consuming.
// ---------------------------------------------------------------------------
__global__ void finalize_kernel(const half_t* __restrict__ vol_v,
                                const half_t* __restrict__ vol_f,
                                const half_t* __restrict__ vol_w,
                                const float*  __restrict__ w_sum,
                                const float*  __restrict__ u_sum,
                                const float*  __restrict__ we_sum,
                                const float*  __restrict__ feat_sum,
                                const u32*    __restrict__ touched,
                                const u32*    __restrict__ touchedE,
                                half_t* __restrict__ out_v,
                                half_t* __restrict__ out_f,
                                half_t* __restrict__ out_w) {
    __shared__ __align__(16) float stage[TPB * 8];
    int i = blockIdx.x * TPB + threadIdx.x;

    // Async-stage this thread's 8-float feat_sum row into LDS.
    // INST_OFFSET is added to BOTH the global and LDS addresses (ISA 08 §4.4),
    // so one base + offset:16 covers the 32-byte row with two B128 transfers.
    unsigned lds_off = (unsigned)(unsigned long long)(&stage[threadIdx.x * 8]);
    const float* fsp = feat_sum + (size_t)i * 8;
    asm volatile("global_load_async_to_lds_b128 %0, %1, off"
                 :: "v"(lds_off), "v"(fsp) : "memory");
    asm volatile("global_load_async_to_lds_b128 %0, %1, off offset:16"
                 :: "v"(lds_off), "v"(fsp) : "memory");

    // Independent scalar work overlaps the async copy.
    bool t  = (touched[i >> 5]  >> (i & 31)) & 1u;
    bool te = (touchedE[i >> 5] >> (i & 31)) & 1u;

    float v_old = (float)vol_v[i];
    float w_old = (float)vol_w[i];
    float ws  = w_sum[i];
    float us  = u_sum[i];
    float wes = we_sum[i];
    v8h f_old = *(const v8h*)(vol_f + (size_t)i * 8);

    float denom = t ? (w_old + ws) : 1.0f;
    float rden  = 1.0f / denom;
    float v_new = t ? (w_old * v_old + us) * rden : v_old;
    float w_new = t ? fminf(fmaxf(w_old + ws, 0.0f), MAX_WEIGHT) : w_old;

    // Empty-space updates are applied AFTER (overwrite), using ORIGINAL v/w.
    if (te) {
        float de = w_old + wes;
        v_new = (w_old * v_old + TRUNC_VALUE * wes) / de;
        w_new = fminf(fmaxf(de, 0.0f), MAX_WEIGHT);
    }

    asm volatile("s_wait_asynccnt 0x0" ::: "memory");

    v8h f_new;
#pragma unroll
    for (int f = 0; f < 8; ++f) {
        float fo = (float)f_old[f];
        float fs = stage[threadIdx.x * 8 + f];
        f_new[f] = (half_t)(t ? (w_old * fo + fs) * rden : fo);
    }

    // Streaming outputs: written once, never re-read -> non-temporal stores.
    __builtin_nontemporal_store((half_t)v_new, out_v + i);
    __builtin_nontemporal_store((half_t)w_new, out_w + i);
    __builtin_nontemporal_store(f_new, (v8h*)(out_f + (size_t)i * 8));
}

// ---------------------------------------------------------------------------
// Launch: zero -> scatter(surface) -> scatter(empty) -> finalize, all on stream
// Scratch layout (f32 words): [w_sum V][u_sum V][we_sum V][feat_sum 8V]
//                             [touched V/32 u32][touchedE V/32 u32]
// ---------------------------------------------------------------------------
extern "C" void kernel_launch(void* const* d_in, const int* in_sizes, int n_in,
                              void* d_out, int out_size, void* d_ws, size_t ws_size,
                              hipStream_t stream) {
    const float*  uv    = (const float*)d_in[0];   // update_values      (N)
    const float*  uf    = (const float*)d_in[1];   // update_features    (N,8)
    const int*    ui    = (const int*)d_in[2];     // update_indices     (N,3)
    const float*  uw    = (const float*)d_in[3];   // update_weights     (N)
    const int*    uie   = (const int*)d_in[4];     // update_indices_empty (N,3)
    const float*  uwe   = (const float*)d_in[5];   // update_weights_empty (N)
    const half_t* vol_v = (const half_t*)d_in[6];  // values_volume  (V) fp16
    const half_t* vol_f = (const half_t*)d_in[7];  // features_volume(V,8) fp16
    const half_t* vol_w = (const half_t*)d_in[8];  // weights_volume (V) fp16

    const int       N  = in_sizes[0];              // 262144
    const int       Ne = in_sizes[5];              // 262144
    const long long V  = (long long)in_sizes[6];   // 16777216 (256^3)

    float* ws_f     = (float*)d_ws;
    float* w_sum    = ws_f;
    float* u_sum    = ws_f + V;
    float* we_sum   = ws_f + 2 * V;
    float* feat_sum = ws_f + 3 * V;
    u32*   touched  = (u32*)(ws_f + 11 * V);
    u32*   touchedE = touched + (V / 32);

    long long n_words = 11 * V + V / 16;           // f32 accum + 2 bitmasks
    long long n_vec4  = (n_words + 3) / 4;

    zero_ws_kernel<<<(unsigned)((n_vec4 + TPB - 1) / TPB), TPB, 0, stream>>>(
        (uint4*)d_ws, n_vec4);

    scatter_surface_kernel<<<(N + TPB - 1) / TPB, TPB, 0, stream>>>(
        uv, uf, ui, uw, w_sum, u_sum, feat_sum, touched, N);

    scatter_empty_kernel<<<(Ne + TPB - 1) / TPB, TPB, 0, stream>>>(
        uie, uwe, we_sum, touchedE, Ne);

    half_t* out = (half_t*)d_out;                  // [v: V][f: 8V][w: V] fp16
    finalize_kernel<<<(unsigned)(V / TPB), TPB, 0, stream>>>(
        vol_v, vol_f, vol_w, w_sum, u_sum, we_sum, feat_sum,
        touched, touchedE, out, out + V, out + 9 * V);
}